// Graph_Learn_687194768237
// MI455X (gfx1250) — compile-verified
//
#include <hip/hip_runtime.h>

#define V 256
#define F 64
#define NSLICE 32            // B*T
#define ROWPAD 68            // 68*4 = 272 B row stride: 16B aligned, bank-conflict-free
#define SMEM_FLOATS (V * ROWPAD + F)
#define SMEM_BYTES  (SMEM_FLOATS * 4)

typedef __attribute__((ext_vector_type(2))) float v2f;
typedef __attribute__((ext_vector_type(8))) float v8f;
typedef int v4i_vec __attribute__((vector_size(16)));   // matches builtin param type

#define AS1 __attribute__((address_space(1)))
#define AS3 __attribute__((address_space(3)))

#if __has_builtin(__builtin_amdgcn_global_load_async_to_lds_b128)
#define HAVE_ASYNC_LDS 1
#endif

// ---------------------------------------------------------------------------
// Kernel 1: one workgroup per (bt, column-block). 8 waves, each wave owns two
// 16x16 (i,j) tiles of the same 16-column block. tmpS tiles are computed in
// WMMA C-layout registers; denom column-sums come from an all-ones WMMA.
// ---------------------------------------------------------------------------
__global__ __launch_bounds__(256) void k_scores(const float* __restrict__ x,
                                                const float* __restrict__ a,
                                                float* __restrict__ out,
                                                float* __restrict__ denom) {
    extern __shared__ float smem[];
    float* s_x = smem;               // V * ROWPAD
    float* s_a = smem + V * ROWPAD;  // F (offset 69632 B, 16B aligned)

    const int tid = threadIdx.x;
    const int bt  = blockIdx.x >> 4;
    const int cb  = blockIdx.x & 15;
    const float* xg = x + (size_t)bt * (V * F);

    // ---- stage x slice (256x64 f32) into LDS, padded rows ----
    {
        const int f4 = tid & 15;   // which float4 of the row
        const int r0 = tid >> 4;   // row group
        for (int k = 0; k < 16; ++k) {
            const int row = r0 + k * 16;
#ifdef HAVE_ASYNC_LDS
            __builtin_amdgcn_global_load_async_to_lds_b128(
                (v4i_vec*)(xg + row * F + f4 * 4),
                (AS3 v4i_vec*)(s_x + row * ROWPAD + f4 * 4), 0, 0);
#else
            const float4 v = *(const float4*)(xg + row * F + f4 * 4);
            *(float4*)(s_x + row * ROWPAD + f4 * 4) = v;
#endif
        }
        if (tid < F) s_a[tid] = a[tid];
#ifdef HAVE_ASYNC_LDS
#if __has_builtin(__builtin_amdgcn_s_wait_asynccnt)
        __builtin_amdgcn_s_wait_asynccnt(0);
#else
        asm volatile("s_wait_asynccnt 0x0" ::: "memory");
#endif
#endif
        __syncthreads();
    }

    const int wv   = tid >> 5;
    const int lane = tid & 31;
    const int n    = lane & 15;          // WMMA C-layout column
    const int mh   = (lane >> 4) << 3;   // row half: 0 or 8
    const int j0   = cb << 4;

    float* og = out + (size_t)bt * (V * V) + j0 + n;

    v8f sumC = {0.f, 0.f, 0.f, 0.f, 0.f, 0.f, 0.f, 0.f};
    const v2f ones = {1.0f, 1.0f};

    const float* xjrow = s_x + (j0 + n) * ROWPAD;

    for (int tt = 0; tt < 2; ++tt) {
        const int i0 = ((wv << 1) + tt) << 4;

        float acc[8] = {0.f, 0.f, 0.f, 0.f, 0.f, 0.f, 0.f, 0.f};
        for (int f4 = 0; f4 < 16; ++f4) {
            const float4 av = *(const float4*)(s_a + f4 * 4);
            const float4 xj = *(const float4*)(xjrow + f4 * 4);
#pragma unroll
            for (int r = 0; r < 8; ++r) {
                const float4 xi = *(const float4*)(s_x + (i0 + mh + r) * ROWPAD + f4 * 4);
                acc[r] = fmaf(av.x, fabsf(xi.x - xj.x), acc[r]);
                acc[r] = fmaf(av.y, fabsf(xi.y - xj.y), acc[r]);
                acc[r] = fmaf(av.z, fabsf(xi.z - xj.z), acc[r]);
                acc[r] = fmaf(av.w, fabsf(xi.w - xj.w), acc[r]);
            }
        }

        float t8[8];
#pragma unroll
        for (int r = 0; r < 8; ++r) t8[r] = __expf(fmaxf(acc[r], 0.0f));

        // tmpS tile out (row-block i0+mh.., column j0+n): coalesced 64B per half-wave
#pragma unroll
        for (int r = 0; r < 8; ++r) og[(size_t)(i0 + mh + r) * V] = t8[r];

        // Ones-matmul reduction: sum the 16 tile rows per column with WMMA.
        // Tile C-layout VGPR pairs are valid 4x16 B operands (A==ones makes the
        // K permutation irrelevant); after 4 accumulations every C row holds
        // the column sums of this tile, accumulated across both tiles.
#pragma unroll
        for (int p = 0; p < 4; ++p) {
            v2f bb = {t8[2 * p], t8[2 * p + 1]};
            sumC = __builtin_amdgcn_wmma_f32_16x16x4_f32(
                false, ones, false, bb, (short)0, sumC, false, false);
        }
    }

    // sumC VGPR0, lanes 0..15 hold denom partial for columns j0+lane
    if (lane < 16) atomicAdd(&denom[bt * V + j0 + lane], sumC[0]);
}

// ---------------------------------------------------------------------------
// Kernel 2: S[bt,i,j] = tmpS[bt,i,j] / denom[bt,i]  (flat float4 pass)
// ---------------------------------------------------------------------------
__global__ __launch_bounds__(256) void k_norm(float* __restrict__ out,
                                              const float* __restrict__ denom) {
    const size_t idx4 = (size_t)blockIdx.x * blockDim.x + threadIdx.x; // float4 units
    const int bt = (int)(idx4 >> 14);       // 16384 float4 per slice
    const int i  = (int)(idx4 >> 6) & 255;  // row index
    float4 v = ((float4*)out)[idx4];
    const float rd = 1.0f / denom[bt * V + i];
    v.x *= rd; v.y *= rd; v.z *= rd; v.w *= rd;
    ((float4*)out)[idx4] = v;
}

// ---------------------------------------------------------------------------
extern "C" void kernel_launch(void* const* d_in, const int* in_sizes, int n_in,
                              void* d_out, int out_size, void* d_ws, size_t ws_size,
                              hipStream_t stream) {
    const float* x = (const float*)d_in[0];  // [4,8,256,64] f32
    const float* a = (const float*)d_in[1];  // [64,1] f32
    float* out   = (float*)d_out;            // [4,8,256,256] f32
    float* denom = (float*)d_ws;             // [32,256] f32 scratch

    (void)hipMemsetAsync(denom, 0, NSLICE * V * sizeof(float), stream);

    (void)hipFuncSetAttribute((const void*)k_scores,
                              hipFuncAttributeMaxDynamicSharedMemorySize, SMEM_BYTES);

    k_scores<<<dim3(NSLICE * 16), dim3(256), SMEM_BYTES, stream>>>(x, a, out, denom);

    const int total4 = NSLICE * V * V / 4;   // 524288 float4
    k_norm<<<dim3(total4 / 256), dim3(256), 0, stream>>>(out, denom);
}